// BiDenseAdjacency_66468913873336
// MI455X (gfx1250) — compile-verified
//
#include <hip/hip_runtime.h>

typedef __attribute__((ext_vector_type(2))) float v2f;
typedef __attribute__((ext_vector_type(8))) float v8f;

#define BATCH 32
#define NDIM  1024
#define FDIM  128

// ---------------------------------------------------------------------------
// WMMA f32 16x16x4 fragment layouts (per CDNA5 ISA):
//  A (16x4):  lanes 0-15 M=lane, lanes 16-31 M=lane-16;
//             VGPR0 = K(2*half+0), VGPR1 = K(2*half+1) -> one b64 load/lane.
//  B (4x16):  lane n = lane&15; VGPR0 = row K=2*half, VGPR1 = row K=2*half+1.
//  C/D(16x16):lane n = lane&15; VGPR r -> row M = r + 8*half.
// ---------------------------------------------------------------------------

// Stage 1: VW[b] = feats[b] (N x F) @ w (F x F).
// One wave computes a 64x16 tile: 4 row sub-tiles share each B-frag of w.
__global__ __launch_bounds__(256) void bda_stage1_vw(
    const float* __restrict__ feats, const float* __restrict__ w,
    float* __restrict__ vw) {
  const int wave = (blockIdx.x * blockDim.x + threadIdx.x) >> 5;
  const int lane = threadIdx.x & 31;
  const int m    = lane & 15;
  const int half = lane >> 4;

  const int tilesPerBatch = (NDIM / 64) * (FDIM / 16); // 16 * 8 = 128
  const int b   = wave / tilesPerBatch;
  const int rem = wave % tilesPerBatch;
  const int i0  = (rem / (FDIM / 16)) * 64;
  const int j0  = (rem % (FDIM / 16)) * 16;

  const float* aP[4];
#pragma unroll
  for (int r = 0; r < 4; ++r)
    aP[r] = feats + ((size_t)b * NDIM + (i0 + 16 * r + m)) * FDIM + 2 * half;
  const float* wCol = w + j0 + m;

  v8f acc[4] = {};
#pragma unroll 4
  for (int k = 0; k < FDIM; k += 4) {
    v2f bf;
    bf.x = wCol[(size_t)(k + 2 * half + 0) * FDIM];
    bf.y = wCol[(size_t)(k + 2 * half + 1) * FDIM];
#pragma unroll
    for (int r = 0; r < 4; ++r) {
      const float2 t = *reinterpret_cast<const float2*>(aP[r] + k);
      v2f a; a.x = t.x; a.y = t.y;
      acc[r] = __builtin_amdgcn_wmma_f32_16x16x4_f32(
          false, a, false, bf, (short)0, acc[r], false, false);
    }
  }

  float* out = vw + (size_t)b * NDIM * FDIM + j0 + m;
#pragma unroll
  for (int r = 0; r < 4; ++r)
#pragma unroll
    for (int q = 0; q < 8; ++q)
      out[(size_t)(i0 + 16 * r + q + 8 * half) * FDIM] = acc[r][q];
}

// Stage 2: Y[b] = VW[b] (N x F) @ feats[b]^T + bias.
// One wave computes a 64x64 tile (4x4 accumulators): per k-step,
// 4 A-frag loads + 4 B-frag gathers feed 16 WMMAs (0.5 loads/WMMA).
// Bias is folded into the accumulator initialization.
__global__ __launch_bounds__(256) void bda_stage2_y(
    const float* __restrict__ vw, const float* __restrict__ feats,
    const float* __restrict__ bias, float* __restrict__ y) {
  const int wave = (blockIdx.x * blockDim.x + threadIdx.x) >> 5;
  const int lane = threadIdx.x & 31;
  const int m    = lane & 15;
  const int half = lane >> 4;

  const int wavesPerBatch = (NDIM / 64) * (NDIM / 64); // 16 * 16 = 256
  const int b   = wave / wavesPerBatch;
  const int rem = wave % wavesPerBatch;
  const int i0  = (rem / (NDIM / 64)) * 64;
  const int j0  = (rem % (NDIM / 64)) * 64;

  const float* aP[4];
  const float* fP[4];
#pragma unroll
  for (int r = 0; r < 4; ++r) {
    aP[r] = vw    + ((size_t)b * NDIM + (i0 + 16 * r + m)) * FDIM + 2 * half;
    fP[r] = feats + ((size_t)b * NDIM + (j0 + 16 * r + m)) * FDIM + 2 * half;
  }

  const float bv = bias[0];
  v8f acc[4][4];
#pragma unroll
  for (int r = 0; r < 4; ++r)
#pragma unroll
    for (int c = 0; c < 4; ++c) {
      v8f z = {bv, bv, bv, bv, bv, bv, bv, bv};
      acc[r][c] = z;
    }

#pragma unroll 2
  for (int k = 0; k < FDIM; k += 4) {
    v2f af[4], bf[4];
#pragma unroll
    for (int r = 0; r < 4; ++r) {
      const float2 ta = *reinterpret_cast<const float2*>(aP[r] + k);
      af[r].x = ta.x; af[r].y = ta.y;
      const float2 tb = *reinterpret_cast<const float2*>(fP[r] + k);
      bf[r].x = tb.x; bf[r].y = tb.y;
    }
#pragma unroll
    for (int r = 0; r < 4; ++r)
#pragma unroll
      for (int c = 0; c < 4; ++c)
        acc[r][c] = __builtin_amdgcn_wmma_f32_16x16x4_f32(
            false, af[r], false, bf[c], (short)0, acc[r][c], false, false);
  }

  float* yB = y + (size_t)b * NDIM * NDIM + j0 + m;
#pragma unroll
  for (int r = 0; r < 4; ++r)
#pragma unroll
    for (int q = 0; q < 8; ++q) {
      const size_t row = (size_t)(i0 + 16 * r + q + 8 * half) * NDIM;
#pragma unroll
      for (int c = 0; c < 4; ++c)
        yB[row + 16 * c] = acc[r][c][q];
    }
}

// Stage 3: copy feats into the second output slot (overwrites the VW scratch).
__global__ __launch_bounds__(256) void bda_copy_feats(
    const float4* __restrict__ src, float4* __restrict__ dst, int n4) {
  const int i = blockIdx.x * blockDim.x + threadIdx.x;
  if (i < n4) dst[i] = src[i];
}

extern "C" void kernel_launch(void* const* d_in, const int* in_sizes, int n_in,
                              void* d_out, int out_size, void* d_ws, size_t ws_size,
                              hipStream_t stream) {
  (void)in_sizes; (void)n_in; (void)out_size; (void)d_ws; (void)ws_size;
  // inputs: [0]=adjMs (unused by reference math), [1]=feats, [2]=w, [3]=b
  const float* feats = (const float*)d_in[1];
  const float* w     = (const float*)d_in[2];
  const float* bias  = (const float*)d_in[3];

  float* y = (float*)d_out;
  const size_t ySize = (size_t)BATCH * NDIM * NDIM; // 33,554,432 floats
  // The feats-copy slot of d_out is exactly B*N*F floats = size of VW.
  // Use it as scratch for VW (stage1 -> stage2), then overwrite with feats.
  float* vw = y + ySize;

  // Stage 1: 32 * 128 waves; 8 waves per 256-thread block.
  {
    const int waves  = BATCH * (NDIM / 64) * (FDIM / 16); // 4096
    bda_stage1_vw<<<waves / 8, 256, 0, stream>>>(feats, w, vw);
  }
  // Stage 2: 32 * 256 waves; 8 waves per block.
  {
    const int waves  = BATCH * (NDIM / 64) * (NDIM / 64); // 8192
    bda_stage2_y<<<waves / 8, 256, 0, stream>>>(vw, feats, bias, y);
  }
  // Stage 3: overwrite scratch slot with the feats copy.
  {
    const int n4 = (BATCH * NDIM * FDIM) / 4; // 1,048,576
    bda_copy_feats<<<(n4 + 255) / 256, 256, 0, stream>>>(
        (const float4*)feats, (float4*)(y + ySize), n4);
  }
}